// NovaGNNEncoder_9792525435307
// MI455X (gfx1250) — compile-verified
//
#include <hip/hip_runtime.h>
#include <hip/hip_bf16.h>

#define N_NODES 50000
#define N_EDGES 800000
#define HIDDEN  128

typedef __attribute__((ext_vector_type(2))) float v2f;
typedef __attribute__((ext_vector_type(8))) float v8f;

// ---------------------------------------------------------------------------
// Zero a float buffer (grid-stride).
// ---------------------------------------------------------------------------
__global__ void zero_f32(float* __restrict__ p, int n) {
    int i = blockIdx.x * blockDim.x + threadIdx.x;
    int stride = gridDim.x * blockDim.x;
    for (; i < n; i += stride) p[i] = 0.0f;
}

// ---------------------------------------------------------------------------
// In-degree via float atomics (deg buffer pre-zeroed).
// ---------------------------------------------------------------------------
__global__ void degree_kernel(const int* __restrict__ dst, float* __restrict__ deg,
                              int n_edges) {
    int e = blockIdx.x * blockDim.x + threadIdx.x;
    if (e < n_edges) atomicAdd(&deg[dst[e]], 1.0f);
}

__global__ void invdeg_kernel(const float* __restrict__ deg, float* __restrict__ invdeg,
                              int n) {
    int i = blockIdx.x * blockDim.x + threadIdx.x;
    if (i < n) invdeg[i] = 1.0f / fmaxf(deg[i], 1.0f);
}

// ---------------------------------------------------------------------------
// Scatter-sum: summed[dst] += x[src]. One thread handles 4 features of one
// edge (float4 gather, 4 f32 atomics). x and summed are L2-resident (25.6MB
// each vs 192MB L2), so atomics resolve in L2.
// ---------------------------------------------------------------------------
__global__ void scatter_add_kernel(const float* __restrict__ xin,
                                   const int* __restrict__ src,
                                   const int* __restrict__ dst,
                                   float* __restrict__ summed, int n_edges) {
    int gid = blockIdx.x * blockDim.x + threadIdx.x;
    int e = gid >> 5;          // 32 threads per edge
    int j = (gid & 31) << 2;   // each covers 4 of 128 features
    if (e >= n_edges) return;
    int s = src[e];
    int d = dst[e];
    const float4 v = *(const float4*)(xin + (size_t)s * HIDDEN + j);
    float* outp = summed + (size_t)d * HIDDEN + j;
    atomicAdd(outp + 0, v.x);
    atomicAdd(outp + 1, v.y);
    atomicAdd(outp + 2, v.z);
    atomicAdd(outp + 3, v.w);
}

// ---------------------------------------------------------------------------
// Fused SAGE linear:  out = (summed * invdeg) @ Wl^T + x @ Wr^T + b  [+ReLU]
// One wave (32 threads) computes a 16-row strip x 128 cols of the output.
// fp32 WMMA 16x16x4, both GEMMs chained through the same C accumulator.
//
// Fragment layouts (wave32, CDNA5 ISA 7.12.2):
//   A 16x4 f32 : lanes 0-15 -> M=lane, VGPR v -> K=v ; lanes 16-31 -> K=v+2
//   B 4x16 f32 : lanes 0-15 -> N=lane, VGPR v -> K=v ; lanes 16-31 -> K=v+2
//   C 16x16 f32: VGPR r -> M=r (lanes 0-15) / M=r+8 (lanes 16-31), N=lane&15
// B[k][n] = W[n][k], so B fragments are contiguous float2 loads from W rows.
// ---------------------------------------------------------------------------
__global__ __launch_bounds__(32) void sage_gemm_kernel(
    const float* __restrict__ summed, const float* __restrict__ invdeg,
    const float* __restrict__ xin,
    const float* __restrict__ Wl, const float* __restrict__ Wr,
    const float* __restrict__ bias,
    float* __restrict__ out, int relu) {
    const int lane  = threadIdx.x;        // 0..31
    const int mlane = lane & 15;
    const int hi    = lane >> 4;          // 0: K=0,1  1: K=2,3
    const int row0  = blockIdx.x * 16;    // 3125 strips * 16 = 50000 exactly
    const int rowA  = row0 + mlane;
    const float inv = invdeg[rowA];

    const float* __restrict__ sumRow = summed + (size_t)rowA * HIDDEN;
    const float* __restrict__ xRow   = xin    + (size_t)rowA * HIDDEN;

    // Accumulators for 8 column tiles, initialized with the bias.
    v8f acc[8];
#pragma unroll
    for (int n = 0; n < 8; ++n) {
        const float bv = bias[n * 16 + mlane];
#pragma unroll
        for (int r = 0; r < 8; ++r) acc[n][r] = bv;
    }

    for (int kk = 0; kk < HIDDEN / 4; ++kk) {
        const int k0 = kk * 4 + hi * 2;   // this lane's K pair

        v2f a_mean = *(const v2f*)(sumRow + k0);
        a_mean[0] *= inv;
        a_mean[1] *= inv;
        v2f a_x = *(const v2f*)(xRow + k0);

#pragma unroll
        for (int n = 0; n < 8; ++n) {
            const int col = n * 16 + mlane;
            v2f b_l = *(const v2f*)(Wl + (size_t)col * HIDDEN + k0);
            v2f b_r = *(const v2f*)(Wr + (size_t)col * HIDDEN + k0);
            acc[n] = __builtin_amdgcn_wmma_f32_16x16x4_f32(
                false, a_mean, false, b_l, (short)0, acc[n], false, false);
            acc[n] = __builtin_amdgcn_wmma_f32_16x16x4_f32(
                false, a_x, false, b_r, (short)0, acc[n], false, false);
        }
    }

    // Store: tile n, VGPR r -> out[row0 + r + 8*hi][n*16 + mlane]
#pragma unroll
    for (int n = 0; n < 8; ++n) {
        const int col = n * 16 + mlane;
#pragma unroll
        for (int r = 0; r < 8; ++r) {
            float v = acc[n][r];
            if (relu) v = fmaxf(v, 0.0f);
            out[(size_t)(row0 + r + 8 * hi) * HIDDEN + col] = v;
        }
    }
}

// ---------------------------------------------------------------------------
extern "C" void kernel_launch(void* const* d_in, const int* in_sizes, int n_in,
                              void* d_out, int out_size, void* d_ws, size_t ws_size,
                              hipStream_t stream) {
    const float* x   = (const float*)d_in[0];
    const int*   ei  = (const int*)d_in[1];   // [2, N_EDGES]
    const float* Wl1 = (const float*)d_in[2];
    const float* Wr1 = (const float*)d_in[3];
    const float* b1  = (const float*)d_in[4];
    const float* Wl2 = (const float*)d_in[5];
    const float* Wr2 = (const float*)d_in[6];
    const float* b2  = (const float*)d_in[7];
    float* out = (float*)d_out;

    const int* src = ei;
    const int* dst = ei + N_EDGES;

    // Workspace layout (floats): summed | deg | invdeg | h1
    float* summed = (float*)d_ws;
    float* deg    = summed + (size_t)N_NODES * HIDDEN;
    float* invdeg = deg + N_NODES;
    float* h1     = invdeg + N_NODES;

    const int ZB = 256;
    const int feat_threads = N_EDGES * 32;          // 4 feats per thread
    const int SC_BLOCKS = (feat_threads + 255) / 256;
    const int ED_BLOCKS = (N_EDGES + 255) / 256;
    const int ND_BLOCKS = (N_NODES + 255) / 256;
    const int STRIPS = N_NODES / 16;                // 3125

    // ---- Layer 1 ----
    zero_f32<<<4096, ZB, 0, stream>>>(summed, N_NODES * HIDDEN + N_NODES); // summed+deg
    degree_kernel<<<ED_BLOCKS, 256, 0, stream>>>(dst, deg, N_EDGES);
    invdeg_kernel<<<ND_BLOCKS, 256, 0, stream>>>(deg, invdeg, N_NODES);
    scatter_add_kernel<<<SC_BLOCKS, 256, 0, stream>>>(x, src, dst, summed, N_EDGES);
    sage_gemm_kernel<<<STRIPS, 32, 0, stream>>>(summed, invdeg, x, Wl1, Wr1, b1, h1, 1);

    // ---- Layer 2 ----
    zero_f32<<<4096, ZB, 0, stream>>>(summed, N_NODES * HIDDEN);
    scatter_add_kernel<<<SC_BLOCKS, 256, 0, stream>>>(h1, src, dst, summed, N_EDGES);
    sage_gemm_kernel<<<STRIPS, 32, 0, stream>>>(summed, invdeg, h1, Wl2, Wr2, b2, out, 0);
}